// GCN_73160472920524
// MI455X (gfx1250) — compile-verified
//
#include <hip/hip_runtime.h>
#include <math.h>

// ---------------------------------------------------------------------------
// GCN forward for MI455X (gfx1250, wave32).
// - Message passing (the memory-bound part) via float4 gathers +
//   global_atomic_add_f32 scatters, always at min(Fin,Fout) width
//   (linearity of segment_sum lets us commute the dense layer).
// - Dense layers via V_WMMA_F32_16X16X4_F32 (exact fp32, 16x16 tiles).
//   One block per 16-row strip; blockDim = 32 * (Fout/16) so the A tile is
//   staged into LDS exactly once and every wave owns one 16-wide N tile.
// ---------------------------------------------------------------------------

#define N_NODES 100000
#define N_EDGES 3200000
#define FMAX    176

typedef __attribute__((ext_vector_type(2))) float v2f;
typedef __attribute__((ext_vector_type(8))) float v8f;

__device__ __forceinline__ float lrelu(float v) { return v > 0.f ? v : 0.01f * v; }

// ---------------- utility kernels ----------------

__global__ void fill_zero(float* __restrict__ p, int n) {
    int i = blockIdx.x * blockDim.x + threadIdx.x;
    if (i < n) p[i] = 0.f;
}

__global__ void degrees_kernel(const int* __restrict__ src, const int* __restrict__ dst,
                               float* __restrict__ degOut, float* __restrict__ degIn, int E) {
    int e = blockIdx.x * blockDim.x + threadIdx.x;
    if (e >= E) return;
    unsafeAtomicAdd(&degOut[src[e]], 1.0f);
    unsafeAtomicAdd(&degIn[dst[e]], 1.0f);
}

__global__ void norms_kernel(const float* __restrict__ degOut, const float* __restrict__ degIn,
                             float* __restrict__ ns, float* __restrict__ nd, int n) {
    int i = blockIdx.x * blockDim.x + threadIdx.x;
    if (i >= n) return;
    float o = degOut[i];
    float d = degIn[i];
    ns[i] = (o > 0.f) ? rsqrtf(fmaxf(o, 1.f)) : 0.f;
    nd[i] = (d > 0.f) ? rsqrtf(fmaxf(d, 1.f)) : 0.f;
}

__global__ void coef_kernel(const int* __restrict__ src, const float* __restrict__ edata,
                            const float* __restrict__ ns, const float* __restrict__ mu,
                            const float* __restrict__ sigma, float* __restrict__ coef, int E) {
    int e = blockIdx.x * blockDim.x + threadIdx.x;
    if (e >= E) return;
    float ed = edata[e];
    float m = *mu, s = *sigma;
    float diff = ed - m;
    float w = (ed == 1.0f) ? ed : expf(-(diff * diff) / s);
    coef[e] = ns[src[e]] * w;
}

// ---------------- message passing: gather + weighted scatter-add ----------------
// thread -> (edge, 4-feature group); feature-fastest for coalesced float4 gathers.
__global__ void scatter_edges(const float* __restrict__ h, const int* __restrict__ src,
                              const int* __restrict__ dst, const float* __restrict__ coef,
                              float* __restrict__ agg, int E, int F) {
    unsigned groups = (unsigned)(F >> 2);             // F is a multiple of 16
    unsigned tid = blockIdx.x * blockDim.x + threadIdx.x;
    unsigned total = (unsigned)E * groups;            // <= 3.2M*40 < 2^31
    if (tid >= total) return;
    unsigned e = tid / groups;
    unsigned g = tid - e * groups;
    int s = src[e], d = dst[e];
    float c = coef[e];
    const float4 hv = *(const float4*)(h + (size_t)s * F + g * 4u);
    float* p = agg + (size_t)d * F + g * 4u;
    unsafeAtomicAdd(p + 0, hv.x * c);
    unsafeAtomicAdd(p + 1, hv.y * c);
    unsafeAtomicAdd(p + 2, hv.z * c);
    unsafeAtomicAdd(p + 3, hv.w * c);
}

// ---------------- fp32 WMMA GEMM: out = [lrelu]((A [*nd]) @ W [+ b]) ----------------
// One 16-row strip of A per block (staged once in padded LDS); blockDim =
// 32 * (Fout/16), each wave owns one 16-wide column tile; K loop in steps of 4
// feeding v_wmma_f32_16x16x4_f32. Wave32 layouts per CDNA5 ISA 7.12.2:
//   A lane(l<16)=row M=l, a.x=K(2*half), a.y=K(2*half+1)
//   B lane(l<16)=col N=l, b.x=W[K=2*half][n], b.y=W[K=2*half+1][n]
//   D vgpr r: M = r + 8*half, N = lane&15
template <bool SCALE_ND, bool EPI>
__global__ void gemm_wmma(const float* __restrict__ A, const float* __restrict__ W,
                          const float* __restrict__ bias, const float* __restrict__ nd,
                          float* __restrict__ out, int M, int Fin, int Fout) {
    extern __shared__ float As[];
    const int LDA = Fin + 4;  // stride % 64 banks != 0 -> conflict-free column reads
    int mBase = blockIdx.x * 16;

    for (int idx = threadIdx.x; idx < 16 * Fin; idx += blockDim.x) {
        int r = idx / Fin;
        int k = idx - r * Fin;
        int m = mBase + r;
        float v = A[(size_t)m * Fin + k];
        if (SCALE_ND) v *= nd[m];
        As[r * LDA + k] = v;
    }
    __syncthreads();

    int wave = threadIdx.x >> 5;
    int lane = threadIdx.x & 31;
    int half = lane >> 4;
    int l16  = lane & 15;
    int nBase = wave * 16;  // blockDim = 32 * (Fout/16): every wave has a tile

    v8f acc = {};
    const float* wcol = W + nBase + l16;
    const float* arow = As + l16 * LDA;
    for (int k0 = 0; k0 < Fin; k0 += 4) {
        int ka = k0 + 2 * half;
        v2f a, b;
        a.x = arow[ka];
        a.y = arow[ka + 1];
        b.x = wcol[(size_t)ka * Fout];
        b.y = wcol[(size_t)(ka + 1) * Fout];
        acc = __builtin_amdgcn_wmma_f32_16x16x4_f32(
            /*neg_a=*/false, a, /*neg_b=*/false, b,
            /*c_mod=*/(short)0, acc, /*reuse_a=*/false, /*reuse_b=*/false);
    }

    int n = nBase + l16;
    float bv = EPI ? bias[n] : 0.f;
#pragma unroll
    for (int r = 0; r < 8; ++r) {
        int m = mBase + r + 8 * half;  // M = 100000 is a multiple of 16: no mask
        float v = acc[r];
        if (EPI) v = lrelu(v + bv);
        out[(size_t)m * Fout + n] = v;
    }
}

// order-B epilogue: h = lrelu(agg * nd[m] + b[f])
__global__ void norm_bias_act(const float* __restrict__ agg, const float* __restrict__ nd,
                              const float* __restrict__ bias, float* __restrict__ out,
                              int M, int F) {
    int i = blockIdx.x * blockDim.x + threadIdx.x;
    if (i >= M * F) return;
    int m = i / F;
    int f = i - m * F;
    out[i] = lrelu(agg[i] * nd[m] + bias[f]);
}

// mean over nodes + lrelu; one feature column per block
__global__ void pool_kernel(const float* __restrict__ h, float* __restrict__ pooled,
                            int M, int F) {
    __shared__ float red[256];
    int f = blockIdx.x;
    float s = 0.f;
    for (int m = threadIdx.x; m < M; m += blockDim.x) s += h[(size_t)m * F + f];
    red[threadIdx.x] = s;
    __syncthreads();
    for (int off = blockDim.x >> 1; off > 0; off >>= 1) {
        if ((int)threadIdx.x < off) red[threadIdx.x] += red[threadIdx.x + off];
        __syncthreads();
    }
    if (threadIdx.x == 0) pooled[f] = lrelu(red[0] / (float)M);
}

// tiny MLP head: 160 -> 140 (lrelu) -> 2 (sigmoid); single block
__global__ void head_kernel(const float* __restrict__ pooled,
                            const float* __restrict__ Wd1, const float* __restrict__ bd1,
                            const float* __restrict__ Wd2, const float* __restrict__ bd2,
                            float* __restrict__ out) {
    __shared__ float p[160];
    __shared__ float h1[140];
    int t = threadIdx.x;
    if (t < 160) p[t] = pooled[t];
    __syncthreads();
    if (t < 140) {
        float s = bd1[t];
        for (int k = 0; k < 160; ++k) s += p[k] * Wd1[k * 140 + t];
        h1[t] = lrelu(s);
    }
    __syncthreads();
    if (t < 2) {
        float s = bd2[t];
        for (int j = 0; j < 140; ++j) s += h1[j] * Wd2[j * 2 + t];
        out[t] = 1.f / (1.f + expf(-s));
    }
}

// ---------------------------------------------------------------------------

extern "C" void kernel_launch(void* const* d_in, const int* in_sizes, int n_in,
                              void* d_out, int out_size, void* d_ws, size_t ws_size,
                              hipStream_t stream) {
    (void)in_sizes; (void)n_in; (void)out_size; (void)ws_size;

    const float* x     = (const float*)d_in[0];
    const int*   src   = (const int*)d_in[1];
    const int*   dst   = (const int*)d_in[2];
    const float* edata = (const float*)d_in[3];
    const float* Wl[11];
    const float* bl[11];
    for (int i = 0; i < 11; ++i) {
        Wl[i] = (const float*)d_in[4 + 2 * i];
        bl[i] = (const float*)d_in[5 + 2 * i];
    }
    const float* Wd1   = (const float*)d_in[26];
    const float* bd1   = (const float*)d_in[27];
    const float* Wd2   = (const float*)d_in[28];
    const float* bd2   = (const float*)d_in[29];
    const float* mu    = (const float*)d_in[30];
    const float* sigma = (const float*)d_in[31];
    float* out = (float*)d_out;

    const int N = N_NODES, E = N_EDGES;

    // carve workspace (256B aligned slots)
    char* ws = (char*)d_ws;
    size_t off = 0;
    auto alloc = [&](size_t nfloats) -> float* {
        float* p = (float*)(ws + off);
        off += ((nfloats * sizeof(float) + 255) / 256) * 256;
        return p;
    };
    float* bufs[3] = { alloc((size_t)N * FMAX), alloc((size_t)N * FMAX), alloc((size_t)N * FMAX) };
    float* coef   = alloc(E);
    float* degO   = alloc(N);
    float* degI   = alloc(N);
    float* ns     = alloc(N);
    float* ndv    = alloc(N);
    float* pooled = alloc(256);

    const int T = 256;
    // ---- graph normalization terms ----
    fill_zero<<<(N + T - 1) / T, T, 0, stream>>>(degO, N);
    fill_zero<<<(N + T - 1) / T, T, 0, stream>>>(degI, N);
    degrees_kernel<<<(E + T - 1) / T, T, 0, stream>>>(src, dst, degO, degI, E);
    norms_kernel<<<(N + T - 1) / T, T, 0, stream>>>(degO, degI, ns, ndv, N);
    coef_kernel<<<(E + T - 1) / T, T, 0, stream>>>(src, edata, ns, mu, sigma, coef, E);

    // ---- 11 GCN layers ----
    const int dims[12] = {64, 80, 160, 112, 160, 176, 96, 144, 96, 128, 96, 160};
    const float* hcur = x;
    int curIdx = -1;  // which of bufs[] holds hcur (-1 = input x)
    for (int l = 0; l < 11; ++l) {
        int Fin = dims[l], Fout = dims[l + 1];
        int i0 = -1, i1 = -1;
        for (int j = 0; j < 3; ++j) {
            if (j == curIdx) continue;
            if (i0 < 0) i0 = j; else i1 = j;
        }
        dim3 gB(N / 16);                         // one block per 16-row strip
        int  tB = 32 * (Fout / 16);              // one wave per 16-wide N tile
        size_t shmem = (size_t)16 * (Fin + 4) * sizeof(float);

        if (Fin <= Fout) {
            // order A: scatter at Fin width, then (agg*nd)@W + b, lrelu
            float* agg = bufs[i0];
            float* nxt = bufs[i1];
            int tot = N * Fin;
            fill_zero<<<(tot + T - 1) / T, T, 0, stream>>>(agg, tot);
            unsigned totE = (unsigned)E * (unsigned)(Fin / 4);
            scatter_edges<<<(totE + T - 1) / T, T, 0, stream>>>(hcur, src, dst, coef, agg, E, Fin);
            gemm_wmma<true, true><<<gB, tB, shmem, stream>>>(agg, Wl[l], bl[l], ndv, nxt, N, Fin, Fout);
            hcur = nxt;
            curIdx = i1;
        } else {
            // order B: h@W first (narrower), scatter at Fout width, then nd+b+lrelu
            float* tmp = bufs[i0];
            float* agg = bufs[i1];
            gemm_wmma<false, false><<<gB, tB, shmem, stream>>>(hcur, Wl[l], nullptr, nullptr, tmp, N, Fin, Fout);
            int tot = N * Fout;
            fill_zero<<<(tot + T - 1) / T, T, 0, stream>>>(agg, tot);
            unsigned totE = (unsigned)E * (unsigned)(Fout / 4);
            scatter_edges<<<(totE + T - 1) / T, T, 0, stream>>>(tmp, src, dst, coef, agg, E, Fout);
            norm_bias_act<<<(tot + T - 1) / T, T, 0, stream>>>(agg, ndv, bl[l], tmp, N, Fout);
            hcur = tmp;
            curIdx = i0;
        }
    }

    // ---- readout ----
    pool_kernel<<<160, 256, 0, stream>>>(hcur, pooled, N, 160);
    head_kernel<<<1, 256, 0, stream>>>(pooled, Wd1, bd1, Wd2, bd2, out);
}